// CoordinateDecoding_71949292142836
// MI455X (gfx1250) — compile-verified
//
#include <hip/hip_runtime.h>
#include <hip/hip_bf16.h>
#include <cstdint>
#include <cstddef>

// Problem geometry (from reference): B=32, C=64, H=W=256
constexpr int HW      = 256 * 256;   // 65536 spatial positions per channel
constexpr int CCH     = 64;          // channels
constexpr int NT      = 256;         // threads per block (8 wave32s)
constexpr int VEC     = 4;           // floats per lane per chunk (b128)
constexpr int CHUNK   = NT * VEC;    // 1024 elements per chunk (4 KB)
constexpr int NCHUNK  = HW / CHUNK;  // 64 chunks per channel
constexpr int DEPTH   = 4;           // async pipeline depth (4 x 4KB LDS staging)

static_assert((NCHUNK - DEPTH) % 4 == 0, "main loop must unroll x4 cleanly");

// ---- CDNA5 async global->LDS copy (ASYNCcnt-tracked DMA-style load) ----
// Each enabled lane copies 16 bytes from the global address gaddr to the LDS address lds_addr.
__device__ __forceinline__ void async_copy_b128(uint32_t lds_addr, const void* gaddr) {
    asm volatile("global_load_async_to_lds_b128 %0, %1, off"
                 :
                 : "v"(lds_addr), "v"(gaddr)
                 : "memory");
}

template <int N>
__device__ __forceinline__ void wait_asynccnt() {
    asm volatile("s_wait_asynccnt %0" : : "i"(N) : "memory");
}

__device__ __forceinline__ void wait_dscnt0() {
    asm volatile("s_wait_dscnt 0" : : : "memory");
}

__global__ void __launch_bounds__(NT)
coord_decode_argmax_kernel(const float* __restrict__ grid,
                           const float* __restrict__ heat,
                           float* __restrict__ out) {
    __shared__ float4 stage[DEPTH][NT];      // 16 KB async staging
    __shared__ float  wave_val[NT / 32];
    __shared__ int    wave_idx[NT / 32];

    const int t  = threadIdx.x;
    const int bc = blockIdx.x;               // 0 .. B*C-1 ; channel slab is contiguous
    const float* __restrict__ src = heat + (size_t)bc * HW + t * VEC;

    // LDS byte offsets of this thread's slot in each staging buffer.
    // (low 32 bits of the flat shared-space pointer == wave-relative LDS address)
    uint32_t lds_slot[DEPTH];
#pragma unroll
    for (int i = 0; i < DEPTH; ++i)
        lds_slot[i] = (uint32_t)(uintptr_t)(&stage[i][t]);

    // ---- prologue: fill the pipeline with chunks 0..DEPTH-1 ----
#pragma unroll
    for (int i = 0; i < DEPTH; ++i)
        async_copy_b128(lds_slot[i], src + (size_t)i * CHUNK);

    float best = -INFINITY;
    int   ibst = 0;

    // Consume staged chunk `k` from buffer `buf` (first-index tie-break).
    auto process = [&](int k, int buf) {
        const float4 v = stage[buf][t];      // ds_load_b128
        const int base = k * CHUNK + t * VEC;
        if (v.x > best) { best = v.x; ibst = base;     }
        if (v.y > best) { best = v.y; ibst = base + 1; }
        if (v.z > best) { best = v.z; ibst = base + 2; }
        if (v.w > best) { best = v.w; ibst = base + 3; }
    };

    // ---- steady state: wait<3>, consume, refill same buffer ----
    // Async loads retire in order, so asynccnt <= 3 with (k+DEPTH) issued
    // guarantees chunk k has landed. s_wait_dscnt 0 closes the unordered
    // DS-read vs async-LDS-write race before the buffer is overwritten.
#pragma unroll 4
    for (int k = 0; k < NCHUNK - DEPTH; ++k) {
        const int buf = k & (DEPTH - 1);
        wait_asynccnt<DEPTH - 1>();
        process(k, buf);
        wait_dscnt0();
        async_copy_b128(lds_slot[buf], src + (size_t)(k + DEPTH) * CHUNK);
    }

    // ---- epilogue: drain the last DEPTH chunks (no more issues) ----
    wait_asynccnt<3>(); process(NCHUNK - 4, (NCHUNK - 4) & (DEPTH - 1));
    wait_asynccnt<2>(); process(NCHUNK - 3, (NCHUNK - 3) & (DEPTH - 1));
    wait_asynccnt<1>(); process(NCHUNK - 2, (NCHUNK - 2) & (DEPTH - 1));
    wait_asynccnt<0>(); process(NCHUNK - 1, (NCHUNK - 1) & (DEPTH - 1));

    // ---- wave32 argmax reduction (first-index tie-break, matches jnp.argmax) ----
#pragma unroll
    for (int m = 16; m > 0; m >>= 1) {
        float ov = __shfl_xor(best, m, 32);
        int   oi = __shfl_xor(ibst, m, 32);
        if (ov > best || (ov == best && oi < ibst)) { best = ov; ibst = oi; }
    }

    const int wid = t >> 5;
    if ((t & 31) == 0) { wave_val[wid] = best; wave_idx[wid] = ibst; }
    __syncthreads();

    if (t == 0) {
        float bb = wave_val[0];
        int   bi = wave_idx[0];
#pragma unroll
        for (int w = 1; w < NT / 32; ++w) {
            float wv = wave_val[w];
            int   wi = wave_idx[w];
            if (wv > bb || (wv == bb && wi < bi)) { bb = wv; bi = wi; }
        }
        const int b = bc / CCH;
        const float* gb = grid + (size_t)b * 2 * HW;
        const float x = gb[bi];        // grid[b, 0, idx]
        const float y = gb[HW + bi];   // grid[b, 1, idx]
        // flip=True: out[b,c,0] = grid[b,1,idx], out[b,c,1] = grid[b,0,idx]
        out[(size_t)bc * 2 + 0] = y;
        out[(size_t)bc * 2 + 1] = x;
    }
}

extern "C" void kernel_launch(void* const* d_in, const int* in_sizes, int n_in,
                              void* d_out, int out_size, void* d_ws, size_t ws_size,
                              hipStream_t stream) {
    (void)in_sizes; (void)n_in; (void)d_ws; (void)ws_size; (void)out_size;
    const float* grid = (const float*)d_in[0];   // [B, 2, H, W] fp32
    const float* heat = (const float*)d_in[1];   // [B, C, H, W] fp32
    float* out = (float*)d_out;                  // [B, C, 2] fp32

    const int nblocks = 32 * CCH;                // one workgroup per (b, c)
    coord_decode_argmax_kernel<<<dim3(nblocks), dim3(NT), 0, stream>>>(grid, heat, out);
}